// ResidualBlock_59665685676166
// MI455X (gfx1250) — compile-verified
//
#include <hip/hip_runtime.h>
#include <math.h>

// ---------------- problem constants ----------------
#define BATCH   2
#define SEQ     2048
#define TOK     (BATCH * SEQ)        // 4096
#define DM      512                  // d_model
#define DIN     1024                 // d_inner
#define NH      16                   // heads
#define HP      64                   // headdim
#define NST     64                   // d_state
#define DXBC    1152                 // d_inner + 2*d_state
#define DPROJ   2192                 // 2*d_inner + 2*d_state + heads

typedef float v2f __attribute__((ext_vector_type(2)));
typedef float v8f __attribute__((ext_vector_type(8)));
typedef int   v4i_gcc __attribute__((vector_size(16)));   // matches builtin's V4i

typedef __attribute__((address_space(1))) v4i_gcc* as1_v4i_p;
typedef __attribute__((address_space(3))) v4i_gcc* as3_v4i_p;
typedef __attribute__((address_space(1))) int*     as1_i32_p;
typedef __attribute__((address_space(3))) int*     as3_i32_p;

__device__ __forceinline__ float sigmoid_f(float x) { return 1.0f / (1.0f + __expf(-x)); }
__device__ __forceinline__ float silu_f(float x)    { return x * sigmoid_f(x); }
__device__ __forceinline__ float softplus_f(float x){ return (x > 20.0f) ? x : log1pf(__expf(x)); }

// ---------------- CDNA5 async global->LDS copy helpers ----------------------
#if __has_builtin(__builtin_amdgcn_global_load_async_to_lds_b128) && \
    __has_builtin(__builtin_amdgcn_global_load_async_to_lds_b32)
#define HAVE_ASYNC_LDS 1
#else
#define HAVE_ASYNC_LDS 0
#endif

__device__ __forceinline__ void async_copy_b128(const float* g, float* l) {
#if HAVE_ASYNC_LDS
    __builtin_amdgcn_global_load_async_to_lds_b128(
        (as1_v4i_p)g, (as3_v4i_p)l, 0, 0);
#else
    *(float4*)l = *(const float4*)g;
#endif
}

__device__ __forceinline__ void async_copy_b32(const float* g, float* l) {
#if HAVE_ASYNC_LDS
    __builtin_amdgcn_global_load_async_to_lds_b32(
        (as1_i32_p)g, (as3_i32_p)l, 0, 0);
#else
    *l = *g;
#endif
}

__device__ __forceinline__ void wait_async0() {
#if HAVE_ASYNC_LDS
#if __has_builtin(__builtin_amdgcn_s_wait_asynccnt)
    __builtin_amdgcn_s_wait_asynccnt(0);
#else
    asm volatile("s_wait_asynccnt 0x0" ::: "memory");
#endif
#endif
}

// ---------------- rmsnorm over 512 (optionally reading flipped sequence) ----
__global__ void rmsnorm512_kernel(const float* __restrict__ x, const float* __restrict__ w,
                                  float* __restrict__ out, int L, int flip) {
    int tok = blockIdx.x;
    int b = tok / L, l = tok % L;
    int sl = flip ? (L - 1 - l) : l;
    const float* row = x + (size_t)(b * L + sl) * DM;
    int t = threadIdx.x;
    float v0 = row[t], v1 = row[t + 256];
    __shared__ float red[256];
    red[t] = v0 * v0 + v1 * v1;
    __syncthreads();
    for (int s = 128; s > 0; s >>= 1) {
        if (t < s) red[t] += red[t + s];
        __syncthreads();
    }
    float scale = rsqrtf(red[0] / (float)DM + 1e-5f);
    float* orow = out + (size_t)tok * DM;
    orow[t]       = v0 * scale * w[t];
    orow[t + 256] = v1 * scale * w[t + 256];
}

// ---------------- tiled fp32 WMMA GEMM with async double-buffered LDS -------
// C = A(MxK) * W + [bias] + [resid]
// WT=0: W row-major (K,N).  WT=1: W row-major (N,K) (i.e. apply W^T).
// Block: 256 threads (8 waves), 64x64 C tile, KT=32 K-slices.
// LDS strides 36/68 floats keep 16B alignment for B128 async writes and
// spread the 16 A-fragment lanes across 16 distinct banks.
template <bool WT>
__global__ void gemm_wmma_f32_kernel(const float* __restrict__ A, const float* __restrict__ W,
                                     const float* __restrict__ bias, const float* __restrict__ resid,
                                     float* __restrict__ C, int M, int N, int K) {
    constexpr int KT = 32;
    __shared__ float As[2][64][36];
    __shared__ float Bs[2][32][68];

    int t    = threadIdx.x;
    int m0   = blockIdx.y * 64;
    int n0   = blockIdx.x * 64;
    int lane = t & 31, wid = t >> 5;
    int wm   = wid & 3, wn = wid >> 2;
    int half = lane >> 4, lm = lane & 15;
    int mo   = wm * 16, no = wn * 32;

    // Boundary N-tile: zero-fill W staging so masked-off async lanes leave 0s.
    if (n0 + 64 > N) {
        for (int e = t; e < 2 * 32 * 68; e += 256) ((float*)Bs)[e] = 0.0f;
    }
    __syncthreads();

    auto issue_tiles = [&](int buf, int k0) {
        // A tile: 64 rows x 32 k, 16B chunks (always in range: M%64==0, K%32==0)
        #pragma unroll
        for (int i = 0; i < 2; i++) {
            int e = t + 256 * i;
            int r = e >> 3, cg = (e & 7) << 2;
            async_copy_b128(&A[(size_t)(m0 + r) * K + k0 + cg], &As[buf][r][cg]);
        }
        if (!WT) {
            // W tile: 32 k-rows x 64 n, contiguous along n -> B128
            #pragma unroll
            for (int i = 0; i < 2; i++) {
                int e = t + 256 * i;
                int r = e >> 4, cg = (e & 15) << 2;
                if (n0 + cg < N)
                    async_copy_b128(&W[(size_t)(k0 + r) * N + n0 + cg], &Bs[buf][r][cg]);
            }
        } else {
            // W^T: contiguous along k in memory but strided in LDS -> B32 gather
            #pragma unroll
            for (int i = 0; i < 8; i++) {
                int e = t + 256 * i;
                int r = e >> 6, c = e & 63;
                if (n0 + c < N)
                    async_copy_b32(&W[(size_t)(n0 + c) * K + k0 + r], &Bs[buf][r][c]);
            }
        }
    };

    v8f acc0 = {};
    v8f acc1 = {};

    int nk = K / KT;
    issue_tiles(0, 0);
    for (int kt = 0; kt < nk; kt++) {
        int buf = kt & 1;
        wait_async0();
        __syncthreads();                       // buf is ready for the whole block
        if (kt + 1 < nk) issue_tiles(buf ^ 1, (kt + 1) * KT);  // overlap next copy

        #pragma unroll
        for (int dk = 0; dk < KT; dk += 4) {
            v2f a;
            a.x = As[buf][mo + lm][dk + 2 * half];
            a.y = As[buf][mo + lm][dk + 2 * half + 1];
            v2f b0;
            b0.x = Bs[buf][dk + 2 * half][no + lm];
            b0.y = Bs[buf][dk + 2 * half + 1][no + lm];
            v2f b1;
            b1.x = Bs[buf][dk + 2 * half][no + 16 + lm];
            b1.y = Bs[buf][dk + 2 * half + 1][no + 16 + lm];
            acc0 = __builtin_amdgcn_wmma_f32_16x16x4_f32(false, a, false, b0, (short)0, acc0, false, false);
            acc1 = __builtin_amdgcn_wmma_f32_16x16x4_f32(false, a, false, b1, (short)0, acc1, false, false);
        }
        __syncthreads();                       // all reads of buf done before overwrite
    }

    #pragma unroll
    for (int r = 0; r < 8; r++) {
        int gm  = m0 + mo + half * 8 + r;
        int gn0 = n0 + no + lm;
        int gn1 = gn0 + 16;
        float v0 = acc0[r], v1 = acc1[r];
        if (bias) {
            if (gn0 < N) v0 += bias[gn0];
            if (gn1 < N) v1 += bias[gn1];
        }
        if (resid) {
            if (gn0 < N) v0 += resid[(size_t)gm * N + gn0];
            if (gn1 < N) v1 += resid[(size_t)gm * N + gn1];
        }
        if (gn0 < N) C[(size_t)gm * N + gn0] = v0;
        if (gn1 < N) C[(size_t)gm * N + gn1] = v1;
    }
}

// ---------------- causal depthwise conv1d (k=4) + SiLU on xBC slice ----------
__global__ void conv_silu_kernel(const float* __restrict__ zx, const float* __restrict__ convw,
                                 const float* __restrict__ convb, float* __restrict__ xbc, int L) {
    int idx = blockIdx.x * blockDim.x + threadIdx.x;
    if (idx >= TOK * DXBC) return;
    int c   = idx % DXBC;
    int tok = idx / DXBC;
    int b = tok / L, l = tok % L;
    float acc = convb[c];
    #pragma unroll
    for (int i = 0; i < 4; i++) {
        int li = l - 3 + i;
        if (li >= 0)
            acc += zx[(size_t)(b * L + li) * DPROJ + DIN + c] * convw[c * 4 + i];
    }
    xbc[(size_t)tok * DXBC + c] = silu_f(acc);
}

// ---------------- dt -> softplus(dt+bias), dA = exp(dt * -exp(Alog)) --------
__global__ void dt_kernel(const float* __restrict__ zx, const float* __restrict__ dtbias,
                          const float* __restrict__ Alog, float* __restrict__ dtt,
                          float* __restrict__ dAv) {
    int idx = blockIdx.x * blockDim.x + threadIdx.x;
    if (idx >= TOK * NH) return;
    int h = idx % NH, tok = idx / NH;
    float raw = zx[(size_t)tok * DPROJ + 2 * DIN + 2 * NST + h];
    float d   = softplus_f(raw + dtbias[h]);
    float A   = -__expf(Alog[h]);
    dtt[idx] = d;
    dAv[idx] = __expf(d * A);
}

// ---------------- selective scan: one block per (batch, head) ---------------
__global__ void scan_kernel(const float* __restrict__ xbc, const float* __restrict__ dtt,
                            const float* __restrict__ dAv, const float* __restrict__ Dp,
                            float* __restrict__ y, int L) {
    int b = blockIdx.x / NH;
    int h = blockIdx.x % NH;
    int t = threadIdx.x;
    int p = t & 63;
    int n0 = (t >> 6) * 16;

    float hreg[16];
    #pragma unroll
    for (int i = 0; i < 16; i++) hreg[i] = 0.0f;

    __shared__ float s_B[64];
    __shared__ float s_C[64];
    __shared__ float s_sc[2];
    __shared__ float s_part[256];

    float dp = Dp[h];

    for (int l = 0; l < L; l++) {
        size_t tok = (size_t)(b * L + l);
        const float* row = xbc + tok * DXBC;
        if (t < 64)        s_B[t]       = row[DIN + t];
        else if (t < 128)  s_C[t - 64]  = row[DIN + NST + (t - 64)];
        else if (t == 128) s_sc[0]      = dAv[tok * NH + h];
        else if (t == 129) s_sc[1]      = dtt[tok * NH + h];
        float xv = row[h * HP + p];
        __syncthreads();

        float a    = s_sc[0];
        float coef = s_sc[1] * xv;
        float acc  = 0.0f;
        #pragma unroll
        for (int i = 0; i < 16; i++) {
            hreg[i] = a * hreg[i] + coef * s_B[n0 + i];
            acc += hreg[i] * s_C[n0 + i];
        }
        s_part[t] = acc;
        __syncthreads();

        if (t < 64) {
            float yv = s_part[t] + s_part[t + 64] + s_part[t + 128] + s_part[t + 192];
            y[tok * DIN + h * HP + t] = yv + dp * xv;   // for t<64, p==t
        }
        __syncthreads();
    }
}

// ---------------- gated rmsnorm over 1024: y = rmsnorm(y * silu(z)) * w -----
__global__ void gatednorm_kernel(float* __restrict__ y, const float* __restrict__ zx,
                                 const float* __restrict__ w) {
    int tok = blockIdx.x;
    int t = threadIdx.x;
    const float* zr = zx + (size_t)tok * DPROJ;   // z = first DIN columns
    float* yr = y + (size_t)tok * DIN;
    float g[4];
    float ss = 0.0f;
    #pragma unroll
    for (int j = 0; j < 4; j++) {
        int c = t + 256 * j;
        float gv = yr[c] * silu_f(zr[c]);
        g[j] = gv;
        ss += gv * gv;
    }
    __shared__ float red[256];
    red[t] = ss;
    __syncthreads();
    for (int s = 128; s > 0; s >>= 1) {
        if (t < s) red[t] += red[t + s];
        __syncthreads();
    }
    float scale = rsqrtf(red[0] / (float)DIN + 1e-5f);
    #pragma unroll
    for (int j = 0; j < 4; j++) {
        int c = t + 256 * j;
        yr[c] = g[j] * scale * w[c];
    }
}

// ---------------- out = [addin +] rmsnorm(mix[maybe flipped] + x, w) --------
__global__ void resnorm_kernel(const float* __restrict__ mix, const float* __restrict__ x,
                               const float* __restrict__ w, const float* __restrict__ addin,
                               float* __restrict__ out, int L, int flip) {
    int tok = blockIdx.x;
    int b = tok / L, l = tok % L;
    int sl = flip ? (L - 1 - l) : l;
    const float* mr = mix + (size_t)(b * L + sl) * DM;
    const float* xr = x + (size_t)tok * DM;
    int t = threadIdx.x;
    float v0 = mr[t] + xr[t];
    float v1 = mr[t + 256] + xr[t + 256];
    __shared__ float red[256];
    red[t] = v0 * v0 + v1 * v1;
    __syncthreads();
    for (int s = 128; s > 0; s >>= 1) {
        if (t < s) red[t] += red[t + s];
        __syncthreads();
    }
    float scale = rsqrtf(red[0] / (float)DM + 1e-5f);
    float o0 = v0 * scale * w[t];
    float o1 = v1 * scale * w[t + 256];
    if (addin) {
        o0 += addin[(size_t)tok * DM + t];
        o1 += addin[(size_t)tok * DM + t + 256];
    }
    out[(size_t)tok * DM + t]       = o0;
    out[(size_t)tok * DM + t + 256] = o1;
}

// ---------------- launch --------------------------------------------------
extern "C" void kernel_launch(void* const* d_in, const int* in_sizes, int n_in,
                              void* d_out, int out_size, void* d_ws, size_t ws_size,
                              hipStream_t stream) {
    (void)in_sizes; (void)n_in; (void)out_size; (void)ws_size;
    const float* x = (const float*)d_in[0];
    // per-direction param blocks: Win, convw, convb, dtbias, Alog, Dp, normw, Wout
    const float* prm[2][8];
    for (int d = 0; d < 2; d++)
        for (int i = 0; i < 8; i++)
            prm[d][i] = (const float*)d_in[1 + d * 8 + i];
    const float* n1f = (const float*)d_in[17];
    const float* n2f = (const float*)d_in[18];
    const float* n1b = (const float*)d_in[19];
    const float* n2b = (const float*)d_in[20];
    const float* ffw = (const float*)d_in[21];   // (out, in) = (512, 512)
    const float* ffb = (const float*)d_in[22];
    float* out = (float*)d_out;

    float* ws = (float*)d_ws;
    size_t o = 0;
    float* XN   = ws + o; o += (size_t)TOK * DM;     // normalized input (aliased as PREFF)
    float* ZX   = ws + o; o += (size_t)TOK * DPROJ;  // in-proj output
    float* XBC  = ws + o; o += (size_t)TOK * DXBC;   // conv+silu output
    float* DTT  = ws + o; o += (size_t)TOK * NH;
    float* DAV  = ws + o; o += (size_t)TOK * NH;
    float* YB   = ws + o; o += (size_t)TOK * DIN;    // scan output / gated-norm (in place)
    float* MIX  = ws + o; o += (size_t)TOK * DM;     // out-proj output
    float* ACC  = ws + o; o += (size_t)TOK * DM;     // forward branch result
    float* PREFF = XN;                               // reuse: XN dead after last in-proj GEMM

    dim3 blk(256);
    dim3 g_norm(TOK);
    dim3 g_in((DPROJ + 63) / 64, TOK / 64);
    dim3 g_out((DM + 63) / 64, TOK / 64);
    dim3 g_conv((TOK * DXBC + 255) / 256);
    dim3 g_dt((TOK * NH + 255) / 256);
    dim3 g_scan(BATCH * NH);

    for (int d = 0; d < 2; d++) {
        const float* Win    = prm[d][0];
        const float* convw  = prm[d][1];
        const float* convb  = prm[d][2];
        const float* dtbias = prm[d][3];
        const float* Alog   = prm[d][4];
        const float* Dpp    = prm[d][5];
        const float* normw  = prm[d][6];
        const float* Wout   = prm[d][7];
        const float* n1 = (d == 0) ? n1f : n1b;
        const float* n2 = (d == 0) ? n2f : n2b;
        int flip = d;  // backward direction reads the flipped sequence

        rmsnorm512_kernel<<<g_norm, blk, 0, stream>>>(x, n1, XN, SEQ, flip);
        gemm_wmma_f32_kernel<false><<<g_in, blk, 0, stream>>>(XN, Win, nullptr, nullptr,
                                                              ZX, TOK, DPROJ, DM);
        conv_silu_kernel<<<g_conv, blk, 0, stream>>>(ZX, convw, convb, XBC, SEQ);
        dt_kernel<<<g_dt, blk, 0, stream>>>(ZX, dtbias, Alog, DTT, DAV);
        scan_kernel<<<g_scan, blk, 0, stream>>>(XBC, DTT, DAV, Dpp, YB, SEQ);
        gatednorm_kernel<<<g_norm, blk, 0, stream>>>(YB, ZX, normw);
        gemm_wmma_f32_kernel<false><<<g_out, blk, 0, stream>>>(YB, Wout, nullptr, nullptr,
                                                               MIX, TOK, DM, DIN);
        if (d == 0) {
            resnorm_kernel<<<g_norm, blk, 0, stream>>>(MIX, x, n2, nullptr, ACC, SEQ, 0);
        } else {
            resnorm_kernel<<<g_norm, blk, 0, stream>>>(MIX, x, n2, ACC, PREFF, SEQ, 1);
        }
    }

    // out = (PREFF @ ffw^T + ffb) + PREFF   (ffw stored (out,in) -> WT=true)
    gemm_wmma_f32_kernel<true><<<g_out, blk, 0, stream>>>(PREFF, ffw, ffb, PREFF,
                                                          out, TOK, DM, DM);
}